// GATsimple_12077448036414
// MI455X (gfx1250) — compile-verified
//
#include <hip/hip_runtime.h>
#include <math.h>

typedef __attribute__((ext_vector_type(2))) float v2f;
typedef __attribute__((ext_vector_type(8))) float v8f;

#define NEG_SLOPE 0.2f
#define EPSV 1e-16f

// ---------------------------------------------------------------------------
// GEMM: H[n,64] = act(X[n,64]) @ W[64,64]   (act = optional ReLU on load)
// One wave computes a 16-row x 64-col tile with v_wmma_f32_16x16x4_f32.
// Block = 256 threads = 8 waves = 128 rows.
// W staged in LDS as k-pair interleaved float2 so each B fragment is one
// ds_load_b64: sW[p*64 + c] = { W[2p][c], W[2p+1][c] }.
//
// fp32 WMMA VGPR layouts (CDNA5 ISA 7.12.2):
//   A 16x4 : lanes 0-15 hold M=lane, K={0,1} in v0,v1; lanes 16-31 K={2,3}
//   B 4x16 : lanes 0-15 hold N=lane, rows K={0,1} in v0,v1; lanes 16-31 K={2,3}
//   C 16x16: vj holds M=j (lanes 0-15, N=lane) and M=j+8 (lanes 16-31, N=lane-16)
// ---------------------------------------------------------------------------
template <int RELU>
__global__ __launch_bounds__(256) void gat_gemm_wmma(
    const float* __restrict__ X, const float* __restrict__ W,
    float* __restrict__ H, int n)
{
    __shared__ float2 sW[32 * 64];
    for (int i = threadIdx.x; i < 32 * 64; i += 256) {
        int p = i >> 6;   // k-pair 0..31
        int c = i & 63;   // col
        sW[i] = make_float2(W[(2 * p) * 64 + c], W[(2 * p + 1) * 64 + c]);
    }
    __syncthreads();

    const int wave = threadIdx.x >> 5;
    const int lane = threadIdx.x & 31;
    const int m    = lane & 15;   // row (A) / col (B/C) within 16
    const int half = lane >> 4;   // k-pair select
    const int r0   = blockIdx.x * 128 + wave * 16;

    const int arow = r0 + m;
    const float2* xrow =
        (const float2*)(X + (size_t)((arow < n) ? arow : 0) * 64);

    v8f acc[4] = {};

    for (int kb = 0; kb < 16; ++kb) {
        const int p = kb * 2 + half;      // k-pair index = k0/2
        float2 av = xrow[p];
        if (RELU) { av.x = fmaxf(av.x, 0.0f); av.y = fmaxf(av.y, 0.0f); }
        v2f A;
        A[0] = av.x; A[1] = av.y;
#pragma unroll
        for (int nb = 0; nb < 4; ++nb) {
            float2 bv = sW[p * 64 + nb * 16 + m];
            v2f B;
            B[0] = bv.x; B[1] = bv.y;
            acc[nb] = __builtin_amdgcn_wmma_f32_16x16x4_f32(
                false, A, false, B, (short)0, acc[nb], false, false);
        }
    }

    const int rbase = r0 + half * 8;
    if (r0 + 16 <= n) {
        // full tile: unconditional stores
#pragma unroll
        for (int nb = 0; nb < 4; ++nb) {
            const int col = nb * 16 + m;
#pragma unroll
            for (int j = 0; j < 8; ++j)
                H[(size_t)(rbase + j) * 64 + col] = acc[nb][j];
        }
    } else if (r0 < n) {
        // partial tail tile
#pragma unroll
        for (int nb = 0; nb < 4; ++nb) {
            const int col = nb * 16 + m;
#pragma unroll
            for (int j = 0; j < 8; ++j)
                if (rbase + j < n) H[(size_t)(rbase + j) * 64 + col] = acc[nb][j];
        }
    }
}

// ---------------------------------------------------------------------------
// Per-node attention scores: aS[i] = h[i]·att_src, aD[i] = h[i]·att_dst
// ---------------------------------------------------------------------------
__global__ __launch_bounds__(256) void gat_att_scores(
    const float* __restrict__ H, const float* __restrict__ att_s,
    const float* __restrict__ att_d, float* __restrict__ aS,
    float* __restrict__ aD, int n)
{
    int i = blockIdx.x * blockDim.x + threadIdx.x;
    if (i >= n) return;
    const float4* h4 = (const float4*)(H + (size_t)i * 64);
    const float4* s4 = (const float4*)att_s;
    const float4* d4 = (const float4*)att_d;
    float ss = 0.0f, sd = 0.0f;
#pragma unroll
    for (int c = 0; c < 16; ++c) {
        float4 v = h4[c];
        float4 a = s4[c];
        float4 b = d4[c];
        ss += v.x * a.x + v.y * a.y + v.z * a.z + v.w * a.w;
        sd += v.x * b.x + v.y * b.y + v.z * b.z + v.w * b.w;
    }
    aS[i] = ss;
    aD[i] = sd;
}

// mkey = 0 (== -inf in ordered-uint space), denom = 0
__global__ __launch_bounds__(256) void gat_init_nodes(
    unsigned* __restrict__ mkey, float* __restrict__ denom, int n)
{
    int i = blockIdx.x * blockDim.x + threadIdx.x;
    if (i < n) { mkey[i] = 0u; denom[i] = 0.0f; }
}

// out[i,:] = bias  (float4 granularity)
__global__ __launch_bounds__(256) void gat_init_out(
    float* __restrict__ out, const float* __restrict__ bias, int n)
{
    int gid = blockIdx.x * blockDim.x + threadIdx.x;
    if (gid >= n * 16) return;
    ((float4*)out)[gid] = ((const float4*)bias)[gid & 15];
}

__device__ __forceinline__ unsigned f32_to_key(float f) {
    unsigned b = __float_as_uint(f);
    return (b & 0x80000000u) ? ~b : (b | 0x80000000u);
}
__device__ __forceinline__ float key_to_f32(unsigned k) {
    unsigned b = (k & 0x80000000u) ? (k ^ 0x80000000u) : ~k;
    return __uint_as_float(b);
}

// e = leaky_relu(aS[src] + aD[dst]); segment-max over dst via uint atomicMax
__global__ __launch_bounds__(256) void gat_edge_max(
    const float* __restrict__ aS, const float* __restrict__ aD,
    const int* __restrict__ src, const int* __restrict__ dst,
    float* __restrict__ evals, unsigned* __restrict__ mkey, int ne)
{
    int ei = blockIdx.x * blockDim.x + threadIdx.x;
    if (ei >= ne) return;
    int d = dst[ei];
    float v = aS[src[ei]] + aD[d];
    v = (v > 0.0f) ? v : NEG_SLOPE * v;
    evals[ei] = v;
    atomicMax(&mkey[d], f32_to_key(v));
}

// ex = exp(e - max[dst]); segment-sum over dst
__global__ __launch_bounds__(256) void gat_edge_expsum(
    const int* __restrict__ dst, const unsigned* __restrict__ mkey,
    float* __restrict__ evals, float* __restrict__ denom, int ne)
{
    int ei = blockIdx.x * blockDim.x + threadIdx.x;
    if (ei >= ne) return;
    int d = dst[ei];
    float ex = __expf(evals[ei] - key_to_f32(mkey[d]));
    evals[ei] = ex;
    atomicAdd(&denom[d], ex);
}

// out[dst] += (ex/(denom[dst]+eps)) * h[src]; 16 threads per edge (float4 each)
__global__ __launch_bounds__(256) void gat_edge_aggregate(
    const float* __restrict__ H, const float* __restrict__ evals,
    const float* __restrict__ denom, const int* __restrict__ src,
    const int* __restrict__ dst, float* __restrict__ out, int ne)
{
    int gid = blockIdx.x * blockDim.x + threadIdx.x;
    if (gid >= ne * 16) return;
    int ei   = gid >> 4;
    int part = gid & 15;
    int s = src[ei];
    int d = dst[ei];
    float alpha = evals[ei] / (denom[d] + EPSV);
    float4 hv = ((const float4*)(H + (size_t)s * 64))[part];
    float* o = out + (size_t)d * 64 + part * 4;
    atomicAdd(o + 0, alpha * hv.x);
    atomicAdd(o + 1, alpha * hv.y);
    atomicAdd(o + 2, alpha * hv.z);
    atomicAdd(o + 3, alpha * hv.w);
}

// ---------------------------------------------------------------------------
extern "C" void kernel_launch(void* const* d_in, const int* in_sizes, int n_in,
                              void* d_out, int out_size, void* d_ws, size_t ws_size,
                              hipStream_t stream)
{
    const float* x    = (const float*)d_in[0];
    const int*   eidx = (const int*)d_in[1];
    const float* W1   = (const float*)d_in[2];
    const float* as1  = (const float*)d_in[3];
    const float* ad1  = (const float*)d_in[4];
    const float* b1   = (const float*)d_in[5];
    const float* W2   = (const float*)d_in[6];
    const float* as2  = (const float*)d_in[7];
    const float* ad2  = (const float*)d_in[8];
    const float* b2   = (const float*)d_in[9];

    const int n  = in_sizes[0] / 64;   // 100000
    const int ne = in_sizes[1] / 2;    // 1600000
    const int* src = eidx;
    const int* dst = eidx + ne;

    // workspace layout
    float*    hA    = (float*)d_ws;              // [n*64] h of current layer
    float*    hB    = hA + (size_t)n * 64;       // [n*64] aggregation accumulator
    float*    ev    = hB + (size_t)n * 64;       // [ne]   edge scalar (e -> ex)
    float*    aS    = ev + (size_t)ne;           // [n]
    float*    aD    = aS + n;                    // [n]
    unsigned* mkey  = (unsigned*)(aD + n);       // [n]
    float*    denom = (float*)(mkey + n);        // [n]

    const int TB = 256;
    dim3 blk(TB);
    dim3 gGemm((n + 127) / 128);
    dim3 gNode((n + TB - 1) / TB);
    dim3 gOut((n * 16 + TB - 1) / TB);
    dim3 gEdge((ne + TB - 1) / TB);
    dim3 gAgg((ne * 16 + TB - 1) / TB);

    // ---------------- Layer 1 ----------------
    gat_gemm_wmma<0><<<gGemm, blk, 0, stream>>>(x, W1, hA, n);
    gat_att_scores<<<gNode, blk, 0, stream>>>(hA, as1, ad1, aS, aD, n);
    gat_init_nodes<<<gNode, blk, 0, stream>>>(mkey, denom, n);
    gat_init_out<<<gOut, blk, 0, stream>>>(hB, b1, n);
    gat_edge_max<<<gEdge, blk, 0, stream>>>(aS, aD, src, dst, ev, mkey, ne);
    gat_edge_expsum<<<gEdge, blk, 0, stream>>>(dst, mkey, ev, denom, ne);
    gat_edge_aggregate<<<gAgg, blk, 0, stream>>>(hA, ev, denom, src, dst, hB, ne);

    // ---------------- Layer 2 (ReLU applied on GEMM load) ----------------
    gat_gemm_wmma<1><<<gGemm, blk, 0, stream>>>(hB, W2, hA, n);
    gat_att_scores<<<gNode, blk, 0, stream>>>(hA, as2, ad2, aS, aD, n);
    gat_init_nodes<<<gNode, blk, 0, stream>>>(mkey, denom, n);
    gat_init_out<<<gOut, blk, 0, stream>>>((float*)d_out, b2, n);
    gat_edge_max<<<gEdge, blk, 0, stream>>>(aS, aD, src, dst, ev, mkey, ne);
    gat_edge_expsum<<<gEdge, blk, 0, stream>>>(dst, mkey, ev, denom, ne);
    gat_edge_aggregate<<<gAgg, blk, 0, stream>>>(hA, ev, denom, src, dst,
                                                 (float*)d_out, ne);
}